// cheby_conv_609885356837
// MI455X (gfx1250) — compile-verified
//
#include <hip/hip_runtime.h>

// ---------------- problem constants ----------------
#define BATCH  32
#define CIN    64
#define NNODE  170
#define TDIM   64
#define COUT   64
#define KCH    3          // Chebyshev order
#define NP     176        // node dim padded to 11*16
#define LQP    177        // Lq row pitch (177 % 64 == 49, coprime -> conflict-free)
#define WP     193        // W row pitch (193 % 64 == 1 -> conflict-free)
#define YP     257        // Ymat row pitch
#define CBLK   4          // channels per c-block
#define YROWS  (CBLK*KCH) // 12 (c,k) rows per block
#define QTILES 11         // ceil(170/16)

typedef __attribute__((ext_vector_type(2))) float v2f;
typedef __attribute__((ext_vector_type(8))) float v8f;

// ---------------- pre-kernel: L2 = 2*adj@adj - I ----------------
__global__ __launch_bounds__(256)
void cheb_build_l2(const float* __restrict__ adj, float* __restrict__ L2) {
    int idx = blockIdx.x * 256 + threadIdx.x;
    if (idx >= NNODE * NNODE) return;
    int q = idx / NNODE, n = idx % NNODE;
    float acc = 0.f;
    for (int m = 0; m < NNODE; ++m)
        acc = fmaf(adj[q * NNODE + m], adj[m * NNODE + n], acc);
    L2[idx] = 2.f * acc - (q == n ? 1.f : 0.f);
}

// ---------------- fused main kernel ----------------
// grid = (QTILES, BATCH), block = 256 (8 wave32)
__global__ __launch_bounds__(256)
void cheby_fused(const float* __restrict__ x,   // [B,CIN,N,T]
                 const float* __restrict__ W,   // [COUT, KCH*CIN] cols = c*3+k
                 const float* __restrict__ adj, // [N,N]  (= L1)
                 const float* __restrict__ L2,  // [N,N]
                 const float* __restrict__ bias,// [COUT]
                 float* __restrict__ out)       // [B,COUT,N,T]
{
    extern __shared__ float smem[];
    float* Lq = smem;                       // [2][16][LQP]  k=1,2 rows of this q-tile
    float* Wl = Lq + 2 * 16 * LQP;          // [COUT][WP]
    float* Xs = Wl + COUT * WP;             // [CBLK][NP][16]
    float* Ym = Xs + CBLK * NP * 16;        // [YROWS][YP]

    const int tid  = threadIdx.x;
    const int lane = tid & 31;
    const int wave = tid >> 5;
    const int qb   = blockIdx.x;            // q-tile
    const int b    = blockIdx.y;            // batch

    // ---- stage W ----
    for (int i = tid; i < COUT * (KCH * CIN); i += 256) {
        int o = i / (KCH * CIN), j = i % (KCH * CIN);
        Wl[o * WP + j] = W[i];
    }
    // ---- stage Lq rows (zero-padded) ----
    for (int i = tid; i < 2 * 16 * LQP; i += 256) {
        int kk  = i / (16 * LQP);
        int rem = i % (16 * LQP);
        int q = rem / LQP, n = rem % LQP;
        int qg = qb * 16 + q;
        float v = 0.f;
        if (qg < NNODE && n < NNODE)
            v = (kk == 0) ? adj[qg * NNODE + n] : L2[qg * NNODE + n];
        Lq[i] = v;
    }
    // ---- zero the Xs pad rows (n in [170,176)) once; async loads never touch them ----
    for (int i = tid; i < CBLK * (NP - NNODE) * 16; i += 256) {
        int cc = i / ((NP - NNODE) * 16);
        int r  = i % ((NP - NNODE) * 16);
        int n  = NNODE + r / 16;
        Xs[(cc * NP + n) * 16 + (r & 15)] = 0.f;
    }
    __syncthreads();

    // LDS byte address of Xs (aperture addresses carry the LDS offset in addr[31:0])
    const unsigned xsBase = (unsigned)(unsigned long long)(const void*)Xs;
    // per-batch global base (uniform -> SGPR pair for the async saddr form)
    const float* xb = x + (size_t)b * CIN * NNODE * TDIM;

    const int hi   = lane >> 4;             // half-wave
    const int lno  = lane & 15;
    const int koff = hi * 2;                // K offset of this half-wave in A/B frags

    for (int tc = 0; tc < TDIM / 16; ++tc) {
        v8f acc[8] = {};                    // 8 out tiles per wave: [o-tile, q=col-tile]

        for (int cb = 0; cb < CIN / CBLK; ++cb) {
            __syncthreads();
            // ---- async-stage x[b, c-block, 0:170, t-chunk] into Xs: B128 per lane ----
            // 4 ch * 170 rows * 4 xfers(16B) = 2720 transfers, ASYNCcnt-tracked
            for (int i = tid; i < CBLK * NNODE * 4; i += 256) {
                int cc = i / (NNODE * 4);
                int r  = i % (NNODE * 4);
                int n  = r >> 2;
                int t4 = r & 3;
                unsigned ldsa = xsBase + (unsigned)(((cc * NP + n) * 16 + t4 * 4) * 4);
                unsigned voff = (unsigned)((((cb * CBLK + cc) * NNODE + n) * TDIM
                                            + tc * 16 + t4 * 4) * 4);
                asm volatile("global_load_async_to_lds_b128 %0, %1, %2 offset:0"
                             :: "v"(ldsa), "v"(voff), "s"(xb)
                             : "memory");
            }
            asm volatile("s_wait_asynccnt 0" ::: "memory");
            __syncthreads();

            // ---- stage 1: Y_k tiles via f32 WMMA, K-loop over n ----
            {
                int cc = wave >> 1;         // channel within block
                int kk = wave & 1;          // 0 -> cheb k=1 (adj), 1 -> cheb k=2 (L2)
                const float* Xc = Xs + cc * NP * 16;
                const float* Lk = Lq + kk * 16 * LQP;

                if (kk == 0) {              // cheb k=0: identity -> row copy
                    #pragma unroll
                    for (int v = 0; v < 8; ++v) {
                        int q = hi * 8 + v;
                        Ym[(cc * KCH + 0) * YP + q * 16 + lno] =
                            Xc[(qb * 16 + q) * 16 + lno];
                    }
                }
                v8f y = {};
                #pragma unroll 4
                for (int nb = 0; nb < NP / 4; ++nb) {
                    int n = nb * 4 + koff;
                    v2f a, bb;
                    a.x  = Lk[lno * LQP + n];
                    a.y  = Lk[lno * LQP + n + 1];
                    bb.x = Xc[n * 16 + lno];
                    bb.y = Xc[(n + 1) * 16 + lno];
                    y = __builtin_amdgcn_wmma_f32_16x16x4_f32(
                            false, a, false, bb, (short)0, y, false, false);
                }
                #pragma unroll
                for (int v = 0; v < 8; ++v) {
                    int q = hi * 8 + v;
                    Ym[(cc * KCH + 1 + kk) * YP + q * 16 + lno] = y[v];
                }
            }
            __syncthreads();

            // ---- stage 2: acc += W[:, block cols] @ Ymat ----
            {
                int ot  = wave >> 1;              // o-tile 0..3
                int ct0 = (wave & 1) * 8;         // first col-tile (== q within tile)
                int o   = ot * 16 + lno;
                #pragma unroll
                for (int rs = 0; rs < YROWS / 4; ++rs) {
                    v2f a;
                    int j = cb * YROWS + rs * 4 + koff;
                    a.x = Wl[o * WP + j];
                    a.y = Wl[o * WP + j + 1];
                    int r = rs * 4 + koff;
                    #pragma unroll
                    for (int ci = 0; ci < 8; ++ci) {
                        int col = (ct0 + ci) * 16 + lno;
                        v2f bb;
                        bb.x = Ym[r * YP + col];
                        bb.y = Ym[(r + 1) * YP + col];
                        acc[ci] = __builtin_amdgcn_wmma_f32_16x16x4_f32(
                                      false, a, false, bb, (short)0, acc[ci],
                                      false, false);
                    }
                }
            }
        } // c-blocks

        // ---- store: tile (ot, ci) is out[o-tile, q = ct0+ci, t-chunk] ----
        {
            int ot  = wave >> 1;
            int ct0 = (wave & 1) * 8;
            #pragma unroll
            for (int ci = 0; ci < 8; ++ci) {
                int qg = qb * 16 + ct0 + ci;
                if (qg < NNODE) {
                    #pragma unroll
                    for (int v = 0; v < 8; ++v) {
                        int o = ot * 16 + hi * 8 + v;
                        out[(((size_t)b * COUT + o) * NNODE + qg) * TDIM
                            + tc * 16 + lno] = acc[ci][v] + bias[o];
                    }
                }
            }
        }
    } // t-chunks
}

// ---------------- launcher ----------------
extern "C" void kernel_launch(void* const* d_in, const int* in_sizes, int n_in,
                              void* d_out, int out_size, void* d_ws, size_t ws_size,
                              hipStream_t stream) {
    const float* x    = (const float*)d_in[0];
    const float* adj  = (const float*)d_in[1];
    const float* W    = (const float*)d_in[2];
    const float* bias = (const float*)d_in[3];
    // d_in[4] = K (==3), baked into the kernel
    float* L2 = (float*)d_ws;               // N*N floats = 115.6 KB scratch

    cheb_build_l2<<<(NNODE * NNODE + 255) / 256, 256, 0, stream>>>(adj, L2);

    size_t lds = (size_t)(2 * 16 * LQP + COUT * WP + CBLK * NP * 16 + YROWS * YP)
                 * sizeof(float);           // ~129.5 KB
    dim3 grid(QTILES, BATCH);
    cheby_fused<<<grid, 256, lds, stream>>>(x, W, adj, L2, bias, (float*)d_out);
}